// Kmeansfusion_87995289960536
// MI455X (gfx1250) — compile-verified
//
#include <hip/hip_runtime.h>

// ---------------------------------------------------------------------------
// KMeans + fusion for MI455X (gfx1250, wave32).
// Distance "GEMM" (K=3, padded to 4) mapped onto V_WMMA_F32_16X16X4_F32:
//   A row m  = (px, py, pz, 1)
//   B col n  = (-2cx, -2cy, -2cz, |c|^2)
//   D[m][n]  = |c_n|^2 - 2 p_m . c_n      (argmin-equivalent score per row)
// Full sqdist = D + |p_m|^2 (added only where column-argmin needs it).
// ~5 GFLOP total, working set < 192MB L2 -> fuse every reduction into the
// WMMA tile pass; never materialize the 256MB distance matrix.
// All reductions are order-deterministic (no float atomics; u64 atomicMin is
// commutative/associative with unique keys).
// ---------------------------------------------------------------------------

#define NPTS   16384      // N_AGENTS * ANCHOR
#define KC     4096       // number of centers / anchors
#define ADIM_  11
#define FDIM_  256
#define ITERS_ 10

typedef float v2f __attribute__((ext_vector_type(2)));
typedef float v8f __attribute__((ext_vector_type(8)));

// Branchless lexicographic (value, index) min-update. Bitwise |/& (not ||/&&)
// so clang emits straight-line v_cmp + mask ops + v_cndmask with no
// s_and_saveexec / s_cbranch_execz divergence.
__device__ __forceinline__ void lexmin(float& v, int& j, float ov, int oj) {
    bool t = (ov < v) | ((ov == v) & (oj < j));
    v = t ? ov : v;
    j = t ? oj : j;
}

// -------------------- init: pts, norms, centers -----------------------------
__global__ void km_init(const float* __restrict__ ta,
                        float4* __restrict__ pts4, float* __restrict__ pnorm,
                        float4* __restrict__ cen4, float4* __restrict__ cenB,
                        unsigned long long* __restrict__ colmin) {
    int i = blockIdx.x * blockDim.x + threadIdx.x;
    if (i < NPTS) {
        const float* r = ta + (size_t)i * ADIM_;
        float x = r[0], y = r[1], z = r[2];
        pts4[i]  = make_float4(x, y, z, 1.0f);
        pnorm[i] = x * x + y * y + z * z;
    }
    if (i < KC) {
        const float* r = ta + (size_t)(4 * i) * ADIM_;   // step = NPTS/KC = 4
        float x = r[0], y = r[1], z = r[2];
        cen4[i] = make_float4(x, y, z, 0.0f);
        cenB[i] = make_float4(-2.f * x, -2.f * y, -2.f * z, x * x + y * y + z * z);
        colmin[i] = ~0ull;
    }
}

// -------------------- k-means assignment (WMMA) -----------------------------
// 256 blocks x 128 threads; each wave owns one 16-row point tile and scans
// all 256 center tiles; per-lane running min + shfl_xor column reduction.
__global__ void __launch_bounds__(128)
km_assign(const float4* __restrict__ pts4, const float4* __restrict__ cenB,
          int* __restrict__ assign) {
    int lane = threadIdx.x & 31;
    int wave = threadIdx.x >> 5;
    int half = lane >> 4;          // lanes 0-15: K=0,1 / rows 0-7 of C
    int lcol = lane & 15;          // lanes 16-31: K=2,3 / rows 8-15 of C
    int m0 = (blockIdx.x * 4 + wave) * 16;

    float4 p = pts4[m0 + lcol];
    v2f a; a[0] = half ? p.z : p.x; a[1] = half ? p.w : p.y;

    float bestv[8]; int bestj[8];
#pragma unroll
    for (int v = 0; v < 8; ++v) { bestv[v] = __builtin_inff(); bestj[v] = 0; }

#pragma unroll 2
    for (int jt = 0; jt < KC / 16; ++jt) {
        int j0 = jt * 16;
        float4 q = cenB[j0 + lcol];
        v2f b; b[0] = half ? q.z : q.x; b[1] = half ? q.w : q.y;
        v8f c = {};
        c = __builtin_amdgcn_wmma_f32_16x16x4_f32(false, a, false, b,
                                                  (short)0, c, false, false);
        int col = j0 + lcol;
#pragma unroll
        for (int v = 0; v < 8; ++v) {
            bool t = c[v] < bestv[v];            // strict < keeps lowest j
            bestv[v] = t ? c[v] : bestv[v];
            bestj[v] = t ? col  : bestj[v];
        }
    }
    // reduce over the 16 column lanes inside each half-wave (xor<=8 stays in half)
#pragma unroll
    for (int m = 1; m < 16; m <<= 1) {
#pragma unroll
        for (int v = 0; v < 8; ++v) {
            float ov = __shfl_xor(bestv[v], m, 32);
            int   oj = __shfl_xor(bestj[v], m, 32);
            lexmin(bestv[v], bestj[v], ov, oj);
        }
    }
    if (lcol == 0) {
#pragma unroll
        for (int v = 0; v < 8; ++v)
            assign[m0 + v + 8 * half] = bestj[v];
    }
}

// -------------------- deterministic centroid update -------------------------
// One block per center; scan assign[] (64KB, L1-resident) and tree-reduce.
__global__ void __launch_bounds__(128)
km_update(const int* __restrict__ assign, const float4* __restrict__ pts4,
          float4* __restrict__ cen4, float4* __restrict__ cenB) {
    int j = blockIdx.x, tid = threadIdx.x;
    float sx = 0, sy = 0, sz = 0, sc = 0;
    for (int i = tid; i < NPTS; i += 128) {
        if (assign[i] == j) {
            float4 p = pts4[i];
            sx += p.x; sy += p.y; sz += p.z; sc += 1.f;
        }
    }
    __shared__ float rx[128], ry[128], rz[128], rc[128];
    rx[tid] = sx; ry[tid] = sy; rz[tid] = sz; rc[tid] = sc;
    __syncthreads();
    for (int s = 64; s > 0; s >>= 1) {
        if (tid < s) {
            rx[tid] += rx[tid + s]; ry[tid] += ry[tid + s];
            rz[tid] += rz[tid + s]; rc[tid] += rc[tid + s];
        }
        __syncthreads();
    }
    if (tid == 0) {
        float cnt = rc[0];
        float4 c = cen4[j];
        float nx = c.x, ny = c.y, nz = c.z;
        if (cnt > 0.f) { nx = rx[0] / cnt; ny = ry[0] / cnt; nz = rz[0] / cnt; }
        cen4[j] = make_float4(nx, ny, nz, 0.f);
        cenB[j] = make_float4(-2.f * nx, -2.f * ny, -2.f * nz,
                              nx * nx + ny * ny + nz * nz);
    }
}

// -------------------- column argmin (nearest point per proto) ---------------
// One block per 16-proto column tile; B fragment loaded once, rows streamed.
__global__ void __launch_bounds__(128)
km_nearest(const float4* __restrict__ pts4, const float* __restrict__ pnorm,
           const float4* __restrict__ cenB,
           unsigned long long* __restrict__ colmin) {
    int lane = threadIdx.x & 31;
    int wave = threadIdx.x >> 5;
    int half = lane >> 4;
    int lcol = lane & 15;
    int j0 = blockIdx.x * 16;

    float4 q = cenB[j0 + lcol];
    v2f b; b[0] = half ? q.z : q.x; b[1] = half ? q.w : q.y;

    float runv = __builtin_inff(); int runi = 0;
#pragma unroll 2
    for (int rt = wave; rt < NPTS / 16; rt += 4) {
        int m0 = rt * 16;
        float4 p = pts4[m0 + lcol];
        v2f a; a[0] = half ? p.z : p.x; a[1] = half ? p.w : p.y;
        v8f c = {};
        c = __builtin_amdgcn_wmma_f32_16x16x4_f32(false, a, false, b,
                                                  (short)0, c, false, false);
#pragma unroll
        for (int v = 0; v < 8; ++v) {
            int row = m0 + v + 8 * half;
            float s = fmaxf(c[v] + pnorm[row], 0.f);   // full sqdist, clamped like ref
            bool t = s < runv;                         // rows ascend: < keeps lowest idx
            runv = t ? s   : runv;
            runi = t ? row : runi;
        }
    }
    // merge the two half-lanes that share column lcol
    {
        float ov = __shfl_xor(runv, 16, 32);
        int   oi = __shfl_xor(runi, 16, 32);
        lexmin(runv, runi, ov, oi);
    }
    if (half == 0) {
        unsigned long long key =
            ((unsigned long long)__float_as_uint(runv) << 32) | (unsigned)runi;
        atomicMin(&colmin[j0 + lcol], key);   // monotone for fp32 >= 0
    }
}

// -------------------- gather prototypes_full --------------------------------
__global__ void km_protos(const unsigned long long* __restrict__ colmin,
                          const float* __restrict__ ta, float* __restrict__ out) {
    int t = blockIdx.x * blockDim.x + threadIdx.x;
    if (t >= KC * ADIM_) return;
    int j = t / ADIM_, c = t % ADIM_;
    unsigned row = (unsigned)(colmin[j] & 0xffffffffu);
    out[t] = ta[(size_t)row * ADIM_ + c];
}

// -------------------- per-row top-4 + feature fusion ------------------------
// One block per row i in [0,4096); lexicographic (dist, idx) matches top_k ties.
__global__ void __launch_bounds__(128)
km_topk_fused(const float4* __restrict__ pts4, const float* __restrict__ pnorm,
              const float4* __restrict__ cenB, const float* __restrict__ feat,
              float* __restrict__ fused) {
    int i = blockIdx.x, tid = threadIdx.x;
    float4 p = pts4[i];
    float pn = pnorm[i];

    float tv[4]; int tj[4];
#pragma unroll
    for (int m = 0; m < 4; ++m) { tv[m] = __builtin_inff(); tj[m] = 0x7fffffff; }

    for (int j = tid; j < KC; j += 128) {
        float4 q = cenB[j];
        float s = fmaxf(pn + q.x * p.x + q.y * p.y + q.z * p.z + q.w, 0.f);
        if (s < tv[3] || (s == tv[3] && j < tj[3])) {   // rare path: keep the branch
            tv[3] = s; tj[3] = j;
#pragma unroll
            for (int m = 3; m > 0; --m) {
                if (tv[m] < tv[m - 1] || (tv[m] == tv[m - 1] && tj[m] < tj[m - 1])) {
                    float fv = tv[m]; tv[m] = tv[m - 1]; tv[m - 1] = fv;
                    int   fj = tj[m]; tj[m] = tj[m - 1]; tj[m - 1] = fj;
                }
            }
        }
    }

    __shared__ float sv[128 * 4];
    __shared__ int   sj[128 * 4];
#pragma unroll
    for (int m = 0; m < 4; ++m) { sv[tid * 4 + m] = tv[m]; sj[tid * 4 + m] = tj[m]; }
    __syncthreads();

    for (int stride = 64; stride > 0; stride >>= 1) {
        if (tid < stride) {
            float mv[8]; int mj[8];
#pragma unroll
            for (int m = 0; m < 4; ++m) {
                mv[m]     = sv[tid * 4 + m];            mj[m]     = sj[tid * 4 + m];
                mv[4 + m] = sv[(tid + stride) * 4 + m]; mj[4 + m] = sj[(tid + stride) * 4 + m];
            }
            // insertion sort 8 entries, lexicographic (v, j)
            for (int a = 1; a < 8; ++a) {
                float v0 = mv[a]; int q0 = mj[a]; int bpos = a - 1;
                while (bpos >= 0 && (mv[bpos] > v0 || (mv[bpos] == v0 && mj[bpos] > q0))) {
                    mv[bpos + 1] = mv[bpos]; mj[bpos + 1] = mj[bpos]; --bpos;
                }
                mv[bpos + 1] = v0; mj[bpos + 1] = q0;
            }
#pragma unroll
            for (int m = 0; m < 4; ++m) { sv[tid * 4 + m] = mv[m]; sj[tid * 4 + m] = mj[m]; }
        }
        __syncthreads();
    }

    int j0 = sj[0], j1 = sj[1], j2 = sj[2], j3 = sj[3];
    for (int d = tid; d < FDIM_; d += 128) {
        fused[(size_t)i * FDIM_ + d] =
            feat[(size_t)j0 * FDIM_ + d] + feat[(size_t)j1 * FDIM_ + d] +
            feat[(size_t)j2 * FDIM_ + d] + feat[(size_t)j3 * FDIM_ + d];
    }
}

// ---------------------------------------------------------------------------
extern "C" void kernel_launch(void* const* d_in, const int* in_sizes, int n_in,
                              void* d_out, int out_size, void* d_ws, size_t ws_size,
                              hipStream_t stream) {
    const float* ta   = (const float*)d_in[1];   // trans_anchor  (4,4096,11)
    const float* feat = (const float*)d_in[3];   // instance_feature (4,4096,256)

    // workspace layout (total 557056 bytes)
    char* ws = (char*)d_ws;
    float4* pts4  = (float4*)(ws);                         // 16384*16
    float*  pnorm = (float*) (ws + 262144);                // 16384*4
    float4* cen4  = (float4*)(ws + 327680);                // 4096*16
    float4* cenB  = (float4*)(ws + 393216);                // 4096*16
    int*    assign= (int*)   (ws + 458752);                // 16384*4
    unsigned long long* colmin = (unsigned long long*)(ws + 524288); // 4096*8

    float* proto_out = (float*)d_out;                 // 4096*11
    float* fused_out = (float*)d_out + KC * ADIM_;    // 4096*256

    km_init<<<NPTS / 256, 256, 0, stream>>>(ta, pts4, pnorm, cen4, cenB, colmin);

    for (int it = 0; it < ITERS_; ++it) {
        km_assign<<<NPTS / 64, 128, 0, stream>>>(pts4, cenB, assign);
        km_update<<<KC, 128, 0, stream>>>(assign, pts4, cen4, cenB);
    }

    km_nearest<<<KC / 16, 128, 0, stream>>>(pts4, pnorm, cenB, colmin);
    km_protos<<<(KC * ADIM_ + 255) / 256, 256, 0, stream>>>(colmin, ta, proto_out);
    km_topk_fused<<<KC, 128, 0, stream>>>(pts4, pnorm, cenB, feat, fused_out);
}